// MultiBox_Loss_62964220559865
// MI455X (gfx1250) — compile-verified
//
#include <hip/hip_runtime.h>

// ---------------------------------------------------------------------------
// MultiBox loss for MI455X (gfx1250, wave32).
// Latency-bound problem (~3.8MB inputs, ~0.3 GFLOP IoU). Strategy:
//  * never materialize the 10647x2048 IoU matrix (87MB of useless HBM traffic)
//  * stage shared operands into LDS with CDNA5 async global->LDS loads
//    (ASYNCcnt + s_wait_asynccnt) -- the gfx1250-specific data-movement path
//  * deterministic fixed-order reductions (no float atomics)
//  * final 64x4 partial reduction via V_WMMA_F32_16X16X4_F32 with all-ones B
//    (B=ones is layout-independent; A uses the documented 16x4 fp32 layout)
// ---------------------------------------------------------------------------

#define N_A    10647
#define N_GT   2048
#define N_OUTC 85
#define PADB   64      // padded #blocks for partials (real: 42)

typedef float v2f __attribute__((ext_vector_type(2)));
typedef float v8f __attribute__((ext_vector_type(8)));

// ---- CDNA5 async global->LDS (tracked by ASYNCcnt) ------------------------
__device__ __forceinline__ void async_copy_b128(unsigned lds_addr, const void* g) {
  asm volatile("global_load_async_to_lds_b128 %0, %1, off"
               :: "v"(lds_addr), "v"((unsigned long long)g) : "memory");
}
__device__ __forceinline__ void async_copy_b32(unsigned lds_addr, const void* g) {
  asm volatile("global_load_async_to_lds_b32 %0, %1, off"
               :: "v"(lds_addr), "v"((unsigned long long)g) : "memory");
}
__device__ __forceinline__ void wait_async0() {
  asm volatile("s_wait_asynccnt 0" ::: "memory");
}
__device__ __forceinline__ unsigned lds_off(const void* p) {
  return (unsigned)(unsigned long long)p;   // low 32 bits of flat shared addr = LDS offset
}

// ---- helpers --------------------------------------------------------------
__device__ __forceinline__ float iou_c(float ax0,float ay0,float ax1,float ay1,float aA,
                                       float bx0,float by0,float bx1,float by1,float aB) {
  float lx = fmaxf(ax0, bx0), ly = fmaxf(ay0, by0);
  float rx = fminf(ax1, bx1), ry = fminf(ay1, by1);
  float iw = fmaxf(rx - lx, 0.0f), ih = fmaxf(ry - ly, 0.0f);
  float inter = iw * ih;
  return inter / (aA + aB - inter);
}
__device__ __forceinline__ float bce(float x, float t) {
  return fmaxf(x, 0.0f) - x * t + log1pf(expf(-fabsf(x)));
}

// ---- K0: zero flags + partials + counts region ----------------------------
__global__ void k0_init(unsigned* p, int nwords) {
  int i = blockIdx.x * 256 + threadIdx.x;
  if (i < nwords) p[i] = 0u;
}

// ---- K1: per-anchor max/argmax over 2048 GTs; GT boxes async-staged -------
__global__ __launch_bounds__(256) void k1_anchor(
    const float* __restrict__ priors, const float* __restrict__ gt_boxes,
    float* __restrict__ amx_iou, int* __restrict__ amx_idx,
    float* __restrict__ acorn, float* __restrict__ aarea) {
  __shared__ __align__(16) float sG[N_GT * 4];   // gt corners (32KB)
  __shared__ float sArea[N_GT];                  // gt areas   (8KB)
  int tid = threadIdx.x;

  for (int r = 0; r < N_GT / 256; ++r) {         // 8 x b128 per thread
    int j = tid + r * 256;
    async_copy_b128(lds_off(&sG[j * 4]), gt_boxes + (size_t)j * 4);
  }
  wait_async0();
  __syncthreads();

  for (int r = 0; r < N_GT / 256; ++r) {         // cxcywh -> corners + area, in place
    int j = tid + r * 256;
    float cx = sG[j*4+0], cy = sG[j*4+1], w = sG[j*4+2], h = sG[j*4+3];
    float x0 = cx - w * 0.5f, y0 = cy - h * 0.5f;
    float x1 = cx + w * 0.5f, y1 = cy + h * 0.5f;
    sG[j*4+0] = x0; sG[j*4+1] = y0; sG[j*4+2] = x1; sG[j*4+3] = y1;
    sArea[j] = (x1 - x0) * (y1 - y0);
  }
  __syncthreads();

  int i = blockIdx.x * 256 + tid;
  if (i >= N_A) return;
  float cx = priors[(size_t)i*N_OUTC+0], cy = priors[(size_t)i*N_OUTC+1];
  float w  = priors[(size_t)i*N_OUTC+2], h  = priors[(size_t)i*N_OUTC+3];
  float ax0 = cx - w*0.5f, ay0 = cy - h*0.5f, ax1 = cx + w*0.5f, ay1 = cy + h*0.5f;
  float aA = (ax1 - ax0) * (ay1 - ay0);

  float best = -1e30f; int bidx = 0;
  for (int j = 0; j < N_GT; ++j) {               // broadcast LDS reads, conflict-free
    float v = iou_c(ax0,ay0,ax1,ay1,aA, sG[j*4],sG[j*4+1],sG[j*4+2],sG[j*4+3], sArea[j]);
    if (v > best) { best = v; bidx = j; }        // strict > == first-occurrence argmax
  }
  amx_iou[i] = best; amx_idx[i] = bidx;
  acorn[i*4+0] = ax0; acorn[i*4+1] = ay0; acorn[i*4+2] = ax1; acorn[i*4+3] = ay1;
  aarea[i] = aA;
}

// ---- K2: per-GT argmax over anchors; anchor corners async-staged ----------
__global__ __launch_bounds__(256) void k2_gt(
    const float* __restrict__ gt_boxes, const float* __restrict__ acorn,
    const float* __restrict__ aarea, int* __restrict__ gmax_idx,
    int* __restrict__ flags) {
  __shared__ __align__(16) float sC[1024 * 4];   // anchor corners chunk (16KB)
  __shared__ float sA[1024];                     // anchor area chunk (4KB)
  int tid = threadIdx.x;
  int j = blockIdx.x * 256 + tid;                // 8 blocks x 256 = 2048 GTs

  float cx = gt_boxes[j*4+0], cy = gt_boxes[j*4+1];
  float w  = gt_boxes[j*4+2], h  = gt_boxes[j*4+3];
  float bx0 = cx - w*0.5f, by0 = cy - h*0.5f, bx1 = cx + w*0.5f, by1 = cy + h*0.5f;
  float aB = (bx1 - bx0) * (by1 - by0);

  float best = -1e30f; int bidx = 0;
  for (int c0 = 0; c0 < N_A; c0 += 1024) {
    int cnt = min(1024, N_A - c0);
    __syncthreads();                             // LDS reuse fence
    for (int r = 0; r < 4; ++r) {
      int k = tid + r * 256;
      if (k < cnt) {
        async_copy_b128(lds_off(&sC[k * 4]), acorn + (size_t)(c0 + k) * 4);
        async_copy_b32 (lds_off(&sA[k]),     aarea + (size_t)(c0 + k));
      }
    }
    wait_async0();
    __syncthreads();
    for (int k = 0; k < cnt; ++k) {
      float v = iou_c(sC[k*4],sC[k*4+1],sC[k*4+2],sC[k*4+3],sA[k],
                      bx0,by0,bx1,by1,aB);
      if (v > best) { best = v; bidx = c0 + k; }
    }
  }
  gmax_idx[j] = bidx;
  flags[bidx] = 1;                               // benign races: all write 1
}

// ---- K3: label assignment (order matches reference exactly) ---------------
__global__ void k3_labels(const float* __restrict__ amx_iou,
                          const int* __restrict__ flags,
                          float* __restrict__ labels) {
  int i = blockIdx.x * 256 + threadIdx.x;
  if (i >= N_A) return;
  float miou = amx_iou[i];
  float lb = flags[i] ? 1.0f : -1.0f;            // scatter at gt_max_index
  if (miou >= 0.7f) lb = 1.0f;                   // POS_IOU
  if (miou <  0.3f) lb = 0.0f;                   // NEG_IOU (overrides scatter)
  labels[i] = lb;
}

// ---- K4: sampling with demote-first semantics; single-block LDS scan ------
__global__ __launch_bounds__(1024) void k4_sample(float* __restrict__ labels,
                                                  float* __restrict__ counts) {
  __shared__ int sc[1024];
  const int CH = 11;                             // 1024*11 >= 10647
  int t = threadIdx.x;
  int base = t * CH;
  int cnt = N_A - base; cnt = cnt < 0 ? 0 : (cnt > CH ? CH : cnt);

  float lab[CH];
  int cpos = 0;
  for (int k = 0; k < cnt; ++k) { lab[k] = labels[base + k]; cpos += (lab[k] == 1.0f); }

  sc[t] = cpos; __syncthreads();                 // inclusive Hillis-Steele scan
  for (int off = 1; off < 1024; off <<= 1) {
    int v = (t >= off) ? sc[t - off] : 0;
    __syncthreads(); sc[t] += v; __syncthreads();
  }
  int npos = sc[1023];
  int expos = npos - 128; if (expos < 0) expos = 0;
  int run = sc[t] - cpos;                        // exclusive prefix
  for (int k = 0; k < cnt; ++k)
    if (lab[k] == 1.0f) { run++; if (run <= expos) lab[k] = -1.0f; }
  __syncthreads();

  int cneg = 0;
  for (int k = 0; k < cnt; ++k) cneg += (lab[k] == 0.0f);
  sc[t] = cneg; __syncthreads();
  for (int off = 1; off < 1024; off <<= 1) {
    int v = (t >= off) ? sc[t - off] : 0;
    __syncthreads(); sc[t] += v; __syncthreads();
  }
  int nneg = sc[1023];
  int npos_after = npos - expos;                 // == min(npos, 128)
  int exneg = nneg - (256 - npos_after); if (exneg < 0) exneg = 0;
  run = sc[t] - cneg;
  for (int k = 0; k < cnt; ++k)
    if (lab[k] == 0.0f) { run++; if (run <= exneg) lab[k] = -1.0f; }

  for (int k = 0; k < cnt; ++k) labels[base + k] = lab[k];
  if (t == 0) {
    counts[0] = fmaxf((float)npos_after, 1.0f);
    counts[1] = fmaxf((float)(nneg - exneg), 1.0f);
  }
}

// ---- K5: per-anchor loss terms + deterministic LDS tree reduction ---------
__global__ __launch_bounds__(256) void k5_loss(
    const float* __restrict__ pred, const float* __restrict__ gt_boxes,
    const int* __restrict__ gt_labels, const float* __restrict__ labels,
    const int* __restrict__ amx_idx, float* __restrict__ partials) {
  __shared__ float red[4 * 256];
  int tid = threadIdx.x;
  int i = blockIdx.x * 256 + tid;
  float sbox = 0.f, sobj = 0.f, snon = 0.f, scls = 0.f;
  if (i < N_A) {
    float lb = labels[i];
    const float* p = pred + (size_t)i * N_OUTC;
    if (lb == 1.0f) {
      int g = amx_idx[i];
      float d0 = p[0] - gt_boxes[g*4+0], d1 = p[1] - gt_boxes[g*4+1];
      float d2 = p[2] - gt_boxes[g*4+2], d3 = p[3] - gt_boxes[g*4+3];
      sbox = d0*d0 + d1*d1 + d2*d2 + d3*d3;
      sobj = bce(p[4], 1.0f);
      int cls = gt_labels[g] + 5;
      for (int c = 5; c < N_OUTC; ++c) scls += bce(p[c], (c == cls) ? 1.0f : 0.0f);
    } else if (lb == 0.0f) {
      snon = bce(p[4], 0.0f);
    }
  }
  red[tid] = sbox; red[256 + tid] = sobj; red[512 + tid] = snon; red[768 + tid] = scls;
  __syncthreads();
  for (int off = 128; off > 0; off >>= 1) {      // fixed-order tree: deterministic
    if (tid < off)
      for (int q = 0; q < 4; ++q) red[q*256 + tid] += red[q*256 + tid + off];
    __syncthreads();
  }
  if (tid == 0)
    for (int q = 0; q < 4; ++q) partials[blockIdx.x * 4 + q] = red[q * 256];
}

// ---- K6: finalize via V_WMMA_F32_16X16X4_F32 (one wave, EXEC all 1s) ------
// A[m][k] = partials[m+16k][q] (documented 16x4 fp32 A layout), B = ones.
// D[m][*] = sum_k A[m][k]  ->  64 -> 16 values; in-lane sum + shfl_xor(16) finishes.
__global__ __launch_bounds__(32) void k6_final(const float* __restrict__ partials,
                                               const float* __restrict__ counts,
                                               float* __restrict__ out_scalar) {
  int L = threadIdx.x;  // 0..31, all lanes active through the WMMA
  float S[4];
#if __has_builtin(__builtin_amdgcn_wmma_f32_16x16x4_f32)
  v2f bmat; bmat.x = 1.0f; bmat.y = 1.0f;        // all-ones B, layout-independent
  for (int q = 0; q < 4; ++q) {
    int i0 = (L < 16) ? L : (L + 16);            // A[m][k]=partials[m+16k][q]
    int i1 = (L < 16) ? (L + 16) : (L + 32);
    v2f a; a.x = partials[i0 * 4 + q]; a.y = partials[i1 * 4 + q];
    v8f c = {};
    v8f d = __builtin_amdgcn_wmma_f32_16x16x4_f32(
        false, a, false, bmat, (short)0, c, false, false);
    float s = d[0]+d[1]+d[2]+d[3]+d[4]+d[5]+d[6]+d[7];  // rows 0-7 / 8-15 per half
    s += __shfl_xor(s, 16, 32);                  // combine wave halves
    S[q] = s;
  }
#else
  for (int q = 0; q < 4; ++q) {
    float s = 0.f;
    for (int b = 0; b < PADB; ++b) s += partials[b * 4 + q];
    S[q] = s;
  }
#endif
  if (L == 0) {
    float np = counts[0], nn = counts[1];
    *out_scalar = S[0]/np + S[1]/np + S[2]/nn + S[3]/(np * 80.0f);
  }
}

// ---------------------------------------------------------------------------
extern "C" void kernel_launch(void* const* d_in, const int* in_sizes, int n_in,
                              void* d_out, int out_size, void* d_ws, size_t ws_size,
                              hipStream_t stream) {
  (void)in_sizes; (void)n_in; (void)out_size; (void)ws_size;
  const float* pred   = (const float*)d_in[0];
  const float* gtb    = (const float*)d_in[1];
  const int*   gtl    = (const int*)d_in[2];
  const float* priors = (const float*)d_in[3];

  char* ws = (char*)d_ws;                        // 16B-aligned offsets, ~384KB total
  float* amx_iou  = (float*)(ws + 0);            // 42624
  int*   amx_idx  = (int*)  (ws + 42624);        // 42624
  float* acorn    = (float*)(ws + 85248);        // 170496
  float* aarea    = (float*)(ws + 255744);       // 42624
  float* labels   = (float*)(ws + 298368);       // 42624
  int*   gmax     = (int*)  (ws + 340992);       // 8192
  int*   flags    = (int*)  (ws + 349184);       // 42624  <- zeroed
  float* partials = (float*)(ws + 391808);       // 1024   <- zeroed
  float* counts   = (float*)(ws + 392832);       // 32     <- zeroed

  const int NB_A = (N_A + 255) / 256;            // 42
  const int zwords = (392864 - 349184) / 4;      // flags..counts region

  k0_init<<<(zwords + 255) / 256, 256, 0, stream>>>((unsigned*)flags, zwords);
  k1_anchor<<<NB_A, 256, 0, stream>>>(priors, gtb, amx_iou, amx_idx, acorn, aarea);
  k2_gt<<<N_GT / 256, 256, 0, stream>>>(gtb, acorn, aarea, gmax, flags);
  k3_labels<<<NB_A, 256, 0, stream>>>(amx_iou, flags, labels);
  k4_sample<<<1, 1024, 0, stream>>>(labels, counts);

  hipMemcpyAsync(d_out, pred, (size_t)N_A * N_OUTC * sizeof(float),
                 hipMemcpyDeviceToDevice, stream);

  k5_loss<<<NB_A, 256, 0, stream>>>(pred, gtb, gtl, labels, amx_idx, partials);
  k6_final<<<1, 32, 0, stream>>>(partials, counts,
                                 (float*)d_out + (size_t)N_A * N_OUTC);
}